// SinkhornLoss_67740224192626
// MI455X (gfx1250) — compile-verified
//
#include <hip/hip_runtime.h>
#include <math.h>

// Problem constants (match reference: x is [4096, 1024] fp32, EPS = 1.0, 20 iters)
static constexpr int BSZ  = 4096;   // batch (rows of x, dim of K)
static constexpr int EDIM = 1024;   // embedding dim
static constexpr int BT   = 128;    // gram block tile (M and N)
static constexpr int KT   = 32;     // gram k-chunk
static constexpr int LDSS = 36;     // LDS row stride in floats (16B aligned, conflict-free)
static constexpr int NITER = 20;
static constexpr int NCHUNK = EDIM / KT;   // 32

typedef float v2f __attribute__((ext_vector_type(2)));
typedef float v8f __attribute__((ext_vector_type(8)));

#if __has_builtin(__builtin_amdgcn_global_load_async_to_lds_b128) && \
    __has_builtin(__builtin_amdgcn_s_wait_asynccnt)
#define USE_ASYNC_LDS 1
// Toolchain-probed signature: (int4* src, int4* dst, imm offset, imm cpol),
// generic (flat) pointers to 16-byte int vectors.
typedef int v4i_b128 __attribute__((vector_size(16)));
#else
#define USE_ASYNC_LDS 0
#endif

// ---------------------------------------------------------------------------
// sq[i] = sum_e x[i,e]^2          (one row per wave, 8 rows per block)
// ---------------------------------------------------------------------------
__global__ __launch_bounds__(256) void rowsq_kernel(const float* __restrict__ x,
                                                    float* __restrict__ sq) {
  const int t = threadIdx.x, lane = t & 31, w = t >> 5;
  const int row = blockIdx.x * 8 + w;
  const float* xr = x + (size_t)row * EDIM;
  float sum = 0.f;
#pragma unroll
  for (int c = 0; c < EDIM / 128; ++c) {              // 8 chunks of 32 lanes * float4
    const int idx = (c * 32 + lane) * 4;
    float4 v = *(const float4*)&xr[idx];
    sum += v.x * v.x + v.y * v.y + v.z * v.z + v.w * v.w;
  }
#pragma unroll
  for (int off = 16; off > 0; off >>= 1) sum += __shfl_xor(sum, off, 32);
  if (lane == 0) sq[row] = sum;
}

// ---------------------------------------------------------------------------
// a[i] = b[i] = 1
// ---------------------------------------------------------------------------
__global__ __launch_bounds__(256) void init_ones_kernel(float* __restrict__ a,
                                                        float* __restrict__ b) {
  const int i = blockIdx.x * 256 + threadIdx.x;
  if (i < BSZ) { a[i] = 1.0f; b[i] = 1.0f; }
}

// ---------------------------------------------------------------------------
// K[i,j] = exp(-max(sq_i + sq_j - 2*(x_i . x_j), 0))
// Upper-triangular 128x128 blocks only; mirror-write the transposed tile.
// fp32 WMMA 16x16x4; 8 waves per WG, each wave owns a 64x32 sub-tile (4x2 frags).
// Double-buffered LDS: chunk c+1 streams via GLOBAL_LOAD_ASYNC_TO_LDS_B128
// (ASYNCcnt) while WMMAs consume chunk c.
// ---------------------------------------------------------------------------
__global__ __launch_bounds__(256) void gram_exp_kernel(const float* __restrict__ x,
                                                       const float* __restrict__ sq,
                                                       float* __restrict__ Km) {
  const int bi = blockIdx.x;            // row block
  const int bj = blockIdx.y;            // col block
  if (bj < bi) return;                  // uniform exit: symmetry

  __shared__ float As[2][BT * LDSS];    // i-block rows  [2][128][36]
  __shared__ float Bs[2][BT * LDSS];    // j-block rows  [2][128][36]

  const int t    = threadIdx.x;
  const int lane = t & 31;
  const int w    = t >> 5;              // wave 0..7
  const int wm   = (w >> 2) * 64;       // wave M origin: 0 or 64
  const int wn   = (w & 3) * 32;        // wave N origin: 0,32,64,96
  const int half = lane >> 4;           // lane half (ISA fp32 frag layout)
  const int l16  = lane & 15;

  const int i0 = bi * BT;
  const int j0 = bj * BT;

  v8f acc[4][2] = {};                   // 4 M-tiles x 2 N-tiles of 16x16 fp32

  // staging indices: 4 passes, thread t handles (row = p*32 + t/8, col = (t&7)*4)
  const int lr = t >> 3;
  const int lc = (t & 7) * 4;

  // compute on buffer `cur` for one k-chunk (shared by both paths)
  auto compute_chunk = [&](int cur) {
    const float* as = &As[cur][0];
    const float* bs = &Bs[cur][0];
#pragma unroll
    for (int kk = 0; kk < KT / 4; ++kk) {      // 8 k-steps of depth 4
      v2f af[4], bf[2];
      // A 16x4 fp32 frag: lanes 0-15 hold K={0,1}, lanes 16-31 K={2,3} -> one b64/lane
#pragma unroll
      for (int mi = 0; mi < 4; ++mi)
        af[mi] = *(const v2f*)&as[(wm + mi * 16 + l16) * LDSS + kk * 4 + half * 2];
      // B 4x16 fp32 frag: same K split, N striped across lanes
#pragma unroll
      for (int ni = 0; ni < 2; ++ni)
        bf[ni] = *(const v2f*)&bs[(wn + ni * 16 + l16) * LDSS + kk * 4 + half * 2];
#pragma unroll
      for (int mi = 0; mi < 4; ++mi)
#pragma unroll
        for (int ni = 0; ni < 2; ++ni)
          acc[mi][ni] = __builtin_amdgcn_wmma_f32_16x16x4_f32(
              false, af[mi], false, bf[ni], (short)0, acc[mi][ni], false, false);
    }
  };

#if USE_ASYNC_LDS
  // 8 async b128 ops per wave per chunk (4 rows x {A,B}); ASYNCcnt is FIFO per wave.
  auto issue_chunk = [&](int k0, int buf) {
#pragma unroll
    for (int p = 0; p < 4; ++p) {
      const int r = p * 32 + lr;
      float* ga = const_cast<float*>(&x[(size_t)(i0 + r) * EDIM + k0 + lc]);
      float* gb = const_cast<float*>(&x[(size_t)(j0 + r) * EDIM + k0 + lc]);
      __builtin_amdgcn_global_load_async_to_lds_b128(
          (v4i_b128*)ga, (v4i_b128*)&As[buf][r * LDSS + lc], 0, 0);
      __builtin_amdgcn_global_load_async_to_lds_b128(
          (v4i_b128*)gb, (v4i_b128*)&Bs[buf][r * LDSS + lc], 0, 0);
    }
  };

  issue_chunk(0, 0);
  for (int c = 0; c < NCHUNK; ++c) {
    const int cur = c & 1;
    if (c + 1 < NCHUNK) {
      issue_chunk((c + 1) * KT, cur ^ 1);          // stream next chunk into idle buffer
      __builtin_amdgcn_s_wait_asynccnt(8);         // FIFO: chunk c's 8 ops are complete
    } else {
      __builtin_amdgcn_s_wait_asynccnt(0);
    }
    __syncthreads();                               // all waves' chunk-c data visible
    compute_chunk(cur);
    __syncthreads();                               // buf[cur] free to overwrite next iter
  }
#else
  for (int k0 = 0; k0 < EDIM; k0 += KT) {
    __syncthreads();
#pragma unroll
    for (int p = 0; p < 4; ++p) {
      const int r = p * 32 + lr;
      float4 va = *(const float4*)&x[(size_t)(i0 + r) * EDIM + k0 + lc];
      float4 vb = *(const float4*)&x[(size_t)(j0 + r) * EDIM + k0 + lc];
      *(float4*)&As[0][r * LDSS + lc] = va;
      *(float4*)&Bs[0][r * LDSS + lc] = vb;
    }
    __syncthreads();
    compute_chunk(0);
  }
#endif

  // Epilogue: c = max(sq_i + sq_j - 2g, 0); k = exp(-c). C layout: VGPR r -> M = r + 8*half, N = l16.
#pragma unroll
  for (int mi = 0; mi < 4; ++mi) {
#pragma unroll
    for (int ni = 0; ni < 2; ++ni) {
      const int gi0 = i0 + wm + mi * 16 + half * 8;
      const int gj  = j0 + wn + ni * 16 + l16;
      const float sqj = sq[gj];
      float kv[8];
#pragma unroll
      for (int r = 0; r < 8; ++r) {
        const int gi = gi0 + r;
        float c = sq[gi] + sqj - 2.0f * acc[mi][ni][r];
        c = fmaxf(c, 0.0f);
        const float kval = __expf(-c);          // EPS = 1
        kv[r] = kval;
        Km[(size_t)gi * BSZ + gj] = kval;       // coalesced row-major store
      }
      if (bi != bj) {                           // mirrored tile: lane owns 8 consecutive cols
        *(float4*)&Km[(size_t)gj * BSZ + gi0]     = make_float4(kv[0], kv[1], kv[2], kv[3]);
        *(float4*)&Km[(size_t)gj * BSZ + gi0 + 4] = make_float4(kv[4], kv[5], kv[6], kv[7]);
      }
    }
  }
}

// ---------------------------------------------------------------------------
// vout[i] = 1 / sum_j K[i,j] * vin[j]     (one row per wave; vin staged in LDS)
// ---------------------------------------------------------------------------
__global__ __launch_bounds__(256) void matvec_recip_kernel(const float* __restrict__ Km,
                                                           const float* __restrict__ vin,
                                                           float* __restrict__ vout) {
  __shared__ float sb[BSZ];                     // 16 KB
  const int t = threadIdx.x;
#pragma unroll
  for (int p = 0; p < BSZ / 1024; ++p) {        // 4 passes of 256 threads * float4
    const int idx = (p * 256 + t) * 4;
    *(float4*)&sb[idx] = *(const float4*)&vin[idx];
  }
  __syncthreads();

  const int lane = t & 31, w = t >> 5;
  const int row = blockIdx.x * 8 + w;
  const float* kr = Km + (size_t)row * BSZ;
  float sum = 0.f;
#pragma unroll 4
  for (int c = 0; c < BSZ / 128; ++c) {
    const int idx = (c * 32 + lane) * 4;
    float4 kv = *(const float4*)&kr[idx];
    float4 bv = *(const float4*)&sb[idx];
    sum += kv.x * bv.x + kv.y * bv.y + kv.z * bv.z + kv.w * bv.w;
  }
#pragma unroll
  for (int off = 16; off > 0; off >>= 1) sum += __shfl_xor(sum, off, 32);
  if (lane == 0) vout[row] = 1.0f / sum;
}

// ---------------------------------------------------------------------------
// partials[blk] = sum over 8 rows of  a[i] * sum_j (k*c)[i,j] * b[j]
// with c recovered as -log(k); k==0 -> k*c -> 0 (the exact limit).
// ---------------------------------------------------------------------------
__global__ __launch_bounds__(256) void loss_rows_kernel(const float* __restrict__ Km,
                                                        const float* __restrict__ a,
                                                        const float* __restrict__ b,
                                                        float* __restrict__ partials) {
  __shared__ float sb[BSZ];
  __shared__ float wsum[8];
  const int t = threadIdx.x;
#pragma unroll
  for (int p = 0; p < BSZ / 1024; ++p) {
    const int idx = (p * 256 + t) * 4;
    *(float4*)&sb[idx] = *(const float4*)&b[idx];
  }
  __syncthreads();

  const int lane = t & 31, w = t >> 5;
  const int row = blockIdx.x * 8 + w;
  const float* kr = Km + (size_t)row * BSZ;
  float sum = 0.f;
  for (int c = 0; c < BSZ / 128; ++c) {
    const int idx = (c * 32 + lane) * 4;
    float4 kv = *(const float4*)&kr[idx];
    float4 bv = *(const float4*)&sb[idx];
    const float k0 = kv.x, k1 = kv.y, k2 = kv.z, k3 = kv.w;
    sum += (k0 > 0.f ? k0 * (-__logf(k0)) : 0.f) * bv.x;
    sum += (k1 > 0.f ? k1 * (-__logf(k1)) : 0.f) * bv.y;
    sum += (k2 > 0.f ? k2 * (-__logf(k2)) : 0.f) * bv.z;
    sum += (k3 > 0.f ? k3 * (-__logf(k3)) : 0.f) * bv.w;
  }
#pragma unroll
  for (int off = 16; off > 0; off >>= 1) sum += __shfl_xor(sum, off, 32);
  if (lane == 0) wsum[w] = a[row] * sum;
  __syncthreads();
  if (t == 0) {
    float s = 0.f;
#pragma unroll
    for (int i = 0; i < 8; ++i) s += wsum[i];
    partials[blockIdx.x] = s;
  }
}

// ---------------------------------------------------------------------------
// out[0] = sum of 512 partials (deterministic single-block tree reduction)
// ---------------------------------------------------------------------------
__global__ __launch_bounds__(256) void reduce_kernel(const float* __restrict__ partials,
                                                     float* __restrict__ out) {
  __shared__ float s[256];
  const int t = threadIdx.x;
  s[t] = partials[t] + partials[t + 256];
  __syncthreads();
  for (int off = 128; off > 0; off >>= 1) {
    if (t < off) s[t] += s[t + off];
    __syncthreads();
  }
  if (t == 0) out[0] = s[0];
}

// ---------------------------------------------------------------------------
extern "C" void kernel_launch(void* const* d_in, const int* in_sizes, int n_in,
                              void* d_out, int out_size, void* d_ws, size_t ws_size,
                              hipStream_t stream) {
  (void)in_sizes; (void)n_in; (void)out_size; (void)ws_size;
  const float* x = (const float*)d_in[0];
  float* out = (float*)d_out;

  // Workspace layout (floats): K[4096*4096] | sq[4096] | a[4096] | b[4096] | partials[512]
  float* ws = (float*)d_ws;
  float* Km = ws;
  float* sq = Km + (size_t)BSZ * BSZ;
  float* a  = sq + BSZ;
  float* b  = a + BSZ;
  float* partials = b + BSZ;

  rowsq_kernel<<<BSZ / 8, 256, 0, stream>>>(x, sq);
  init_ones_kernel<<<BSZ / 256, 256, 0, stream>>>(a, b);

  dim3 gg(BSZ / BT, BSZ / BT);                        // 32x32, lower triangle exits early
  gram_exp_kernel<<<gg, 256, 0, stream>>>(x, sq, Km);

  for (int it = 0; it < NITER; ++it) {
    matvec_recip_kernel<<<BSZ / 8, 256, 0, stream>>>(Km, b, a);   // a = 1/(K b)
    matvec_recip_kernel<<<BSZ / 8, 256, 0, stream>>>(Km, a, b);   // b = 1/(K a)  (K symmetric)
  }

  loss_rows_kernel<<<BSZ / 8, 256, 0, stream>>>(Km, a, b, partials);
  reduce_kernel<<<1, 256, 0, stream>>>(partials, out);
}